// ChordModel_81106162418459
// MI455X (gfx1250) — compile-verified
//
#include <hip/hip_runtime.h>
#include <stdint.h>

// Problem shape (fixed by setup_inputs): B=32, T=2048, D=512, fp32.
#define B_N 32
#define T_N 2048
#define D_N 512
#define TILE_T 16   // output frames per workgroup in the mean kernel
#define WIN_CAP 32  // max staged window (frames) -> 32*512*4 = 64 KB LDS

typedef unsigned int v4u __attribute__((ext_vector_type(4)));
typedef int v8i __attribute__((ext_vector_type(8)));
typedef int v4i __attribute__((ext_vector_type(4)));

// TDM (tensor_load_to_lds) availability. Arity differs by toolchain:
//   ROCm 7.2 / clang-22:      5 args (g0, g1, g2, g3, cpol)
//   amdgpu-toolchain/clang-23: 6 args (g0, g1, g2, g3, v8i, cpol)
#if defined(__has_builtin)
#if __has_builtin(__builtin_amdgcn_tensor_load_to_lds) && \
    __has_builtin(__builtin_amdgcn_s_wait_tensorcnt)
#define SEGMEAN_HAVE_TDM 1
#endif
#endif

// ---------------------------------------------------------------------------
// Kernel 1: per-row segment bounds.
//   seg_start[t] = largest t' <= t with (t'==0 || cc[t']!=0)
//   seg_end[t]   = smallest t' > t with cc[t']!=0, else T
// One 256-thread block per batch row; each thread owns 8 consecutive frames.
// ---------------------------------------------------------------------------
__global__ __launch_bounds__(256) void seg_bounds_kernel(
    const int* __restrict__ cc, int* __restrict__ segs, int* __restrict__ sege) {
  __shared__ int s_scan[256];
  const int tid = threadIdx.x;
  const int b = blockIdx.x;
  const int* ccb = cc + b * T_N;
  const int base = tid * 8;

  int flag[8];
#pragma unroll
  for (int j = 0; j < 8; ++j) {
    const int t = base + j;
    flag[j] = (t == 0) || (ccb[t] != 0);
  }

  // ---- forward max-scan: last boundary index per chunk, then carry in ----
  int last = -1;
#pragma unroll
  for (int j = 0; j < 8; ++j)
    if (flag[j]) last = base + j;
  s_scan[tid] = last;
  __syncthreads();
  for (int off = 1; off < 256; off <<= 1) {
    const int v = s_scan[tid];
    const int u = (tid >= off) ? s_scan[tid - off] : -1;
    __syncthreads();
    s_scan[tid] = (u > v) ? u : v;
    __syncthreads();
  }
  const int carryS = (tid > 0) ? s_scan[tid - 1] : -1;
  __syncthreads();  // protect s_scan before reuse below

  int runS = carryS;
  int oS[8];
#pragma unroll
  for (int j = 0; j < 8; ++j) {
    const int t = base + j;
    if (flag[j]) runS = t;
    oS[j] = runS;
  }
  {
    int4* sp = (int4*)(segs + b * T_N + base);
    sp[0] = make_int4(oS[0], oS[1], oS[2], oS[3]);
    sp[1] = make_int4(oS[4], oS[5], oS[6], oS[7]);
  }

  // ---- reverse min-scan: first boundary index per chunk, carry from right --
  int first = 0x7FFFFFFF;
#pragma unroll
  for (int j = 7; j >= 0; --j)
    if (flag[j]) first = base + j;
  s_scan[tid] = first;
  __syncthreads();
  for (int off = 1; off < 256; off <<= 1) {
    const int v = s_scan[tid];
    const int u = (tid + off < 256) ? s_scan[tid + off] : 0x7FFFFFFF;
    __syncthreads();
    s_scan[tid] = (u < v) ? u : v;
    __syncthreads();
  }
  int carryE = (tid < 255) ? s_scan[tid + 1] : 0x7FFFFFFF;
  if (carryE > T_N) carryE = T_N;

  int runE = carryE;
  int oE[8];
#pragma unroll
  for (int j = 7; j >= 0; --j) {
    const int t = base + j;
    oE[j] = runE;  // strictly-after boundary
    if (flag[j]) runE = t;
  }
  {
    int4* ep = (int4*)(sege + b * T_N + base);
    ep[0] = make_int4(oE[0], oE[1], oE[2], oE[3]);
    ep[1] = make_int4(oE[4], oE[5], oE[6], oE[7]);
  }
}

// ---------------------------------------------------------------------------
// Kernel 2: segment means via a TDM-staged LDS window.
// Block = (b, 16-frame tile), 256 threads (8 waves). The tile's frames all
// live in window [mn, mx); wave 0 DMAs it into LDS with tensor_load_to_lds,
// then each wave sums its 2 frames' segments from LDS (ds_load_b128) and
// stores coalesced float4 results.
// ---------------------------------------------------------------------------
__global__ __launch_bounds__(256) void seg_mean_kernel(
    const float* __restrict__ H, const int* __restrict__ segs,
    const int* __restrict__ sege, float* __restrict__ out) {
  __shared__ float smem[WIN_CAP * D_N];  // 64 KB staging window
  const int tid = threadIdx.x;
  const int b = blockIdx.y;
  const int t0 = blockIdx.x * TILE_T;
  const int* srow = segs + b * T_N;
  const int* erow = sege + b * T_N;
  const int mn = srow[t0];               // earliest needed frame
  const int mx = erow[t0 + TILE_T - 1];  // one past latest needed frame
  const int win = mx - mn;               // >= TILE_T, workgroup-uniform
  const float* Hrow = H + (size_t)b * T_N * D_N;
  const bool useLds = (win <= WIN_CAP);

  if (useLds) {
    const unsigned nelem = (unsigned)(win * D_N);  // <= 16384 fp32 elements
#ifdef SEGMEAN_HAVE_TDM
    if (tid < 32) {  // wave 0 issues one TDM DMA for the whole window
      const unsigned lds_base = (unsigned)(size_t)(&smem[0]);  // LDS offset
      const unsigned long long ga =
          (unsigned long long)(size_t)(Hrow + (size_t)mn * D_N);
      // D# group 0: count=1 | lds_addr | global_addr[56:0] | type=2
      v4u g0;
      g0[0] = 1u;
      g0[1] = lds_base;
      g0[2] = (unsigned)(ga & 0xFFFFFFFFull);
      g0[3] = (unsigned)((ga >> 32) & 0x01FFFFFFull) | 0x80000000u;
      // D# group 1: data_size=4B, 1-D tile of nelem elements
      v8i g1;
      g1[0] = (int)(2u << 16);                  // data_size=2 (4 bytes)
      g1[1] = (int)((nelem & 0xFFFFu) << 16);   // tensor_dim0[15:0]
      g1[2] = (int)((nelem >> 16) | (1u << 16));// tensor_dim0[31:16]|tensor_dim1=1
      g1[3] = (int)((nelem & 0xFFFFu) << 16);   // tile_dim0 = nelem
      g1[4] = 1;                                // tile_dim1 = 1, tile_dim2 = 0
      g1[5] = (int)nelem;                       // tensor_dim0_stride
      g1[6] = 0;
      g1[7] = 0;
      v4i g2 = {0, 0, 0, 0};
      v4i g3 = {0, 0, 0, 0};
#if defined(__clang_major__) && (__clang_major__ >= 23)
      v8i gx = {0, 0, 0, 0, 0, 0, 0, 0};
      __builtin_amdgcn_tensor_load_to_lds(g0, g1, g2, g3, gx, 0);
#else
      __builtin_amdgcn_tensor_load_to_lds(g0, g1, g2, g3, 0);
#endif
      __builtin_amdgcn_s_wait_tensorcnt(0);
    }
#else
    // Portable fallback: cooperative float4 staging.
    const float4* gsrc = (const float4*)(Hrow + (size_t)mn * D_N);
    for (unsigned i = tid; i < nelem / 4u; i += 256u) {
      ((float4*)smem)[i] = gsrc[i];
    }
#endif
    __syncthreads();  // uniform: useLds is workgroup-uniform
  }

  const int wave = tid >> 5;
  const int lane = tid & 31;
#pragma unroll
  for (int fi = 0; fi < TILE_T / 8; ++fi) {  // 2 frames per wave
    const int t = t0 + wave * (TILE_T / 8) + fi;
    const int s = srow[t];
    const int e = erow[t];
    const float len = (float)(e - s);
    float4 a0 = make_float4(0.f, 0.f, 0.f, 0.f);
    float4 a1 = make_float4(0.f, 0.f, 0.f, 0.f);
    float4 a2 = make_float4(0.f, 0.f, 0.f, 0.f);
    float4 a3 = make_float4(0.f, 0.f, 0.f, 0.f);

    if (useLds) {
      const float* p = &smem[(size_t)(s - mn) * D_N + lane * 4];
      for (int tt = s; tt < e; ++tt) {
        const float4 v0 = *(const float4*)(p);
        const float4 v1 = *(const float4*)(p + 128);
        const float4 v2 = *(const float4*)(p + 256);
        const float4 v3 = *(const float4*)(p + 384);
        a0.x += v0.x; a0.y += v0.y; a0.z += v0.z; a0.w += v0.w;
        a1.x += v1.x; a1.y += v1.y; a1.z += v1.z; a1.w += v1.w;
        a2.x += v2.x; a2.y += v2.y; a2.z += v2.z; a2.w += v2.w;
        a3.x += v3.x; a3.y += v3.y; a3.z += v3.z; a3.w += v3.w;
        p += D_N;
      }
    } else {
      const float* p = Hrow + (size_t)s * D_N + lane * 4;
      for (int tt = s; tt < e; ++tt) {
        const float4 v0 = *(const float4*)(p);
        const float4 v1 = *(const float4*)(p + 128);
        const float4 v2 = *(const float4*)(p + 256);
        const float4 v3 = *(const float4*)(p + 384);
        a0.x += v0.x; a0.y += v0.y; a0.z += v0.z; a0.w += v0.w;
        a1.x += v1.x; a1.y += v1.y; a1.z += v1.z; a1.w += v1.w;
        a2.x += v2.x; a2.y += v2.y; a2.z += v2.z; a2.w += v2.w;
        a3.x += v3.x; a3.y += v3.y; a3.z += v3.z; a3.w += v3.w;
        p += D_N;
      }
    }

    float* o = out + ((size_t)b * T_N + t) * D_N + lane * 4;
    *(float4*)(o)       = make_float4(a0.x / len, a0.y / len, a0.z / len, a0.w / len);
    *(float4*)(o + 128) = make_float4(a1.x / len, a1.y / len, a1.z / len, a1.w / len);
    *(float4*)(o + 256) = make_float4(a2.x / len, a2.y / len, a2.z / len, a2.w / len);
    *(float4*)(o + 384) = make_float4(a3.x / len, a3.y / len, a3.z / len, a3.w / len);
  }
}

extern "C" void kernel_launch(void* const* d_in, const int* in_sizes, int n_in,
                              void* d_out, int out_size, void* d_ws, size_t ws_size,
                              hipStream_t stream) {
  (void)in_sizes; (void)n_in; (void)out_size; (void)ws_size;
  const float* H = (const float*)d_in[0];  // hidden_states [32,2048,512] f32
  const int* cc = (const int*)d_in[1];     // chord_changes [32,2048] i32
  float* out = (float*)d_out;

  int* segs = (int*)d_ws;                  // [32,2048] seg_start (256 KB)
  int* sege = segs + B_N * T_N;            // [32,2048] seg_end   (256 KB)

  seg_bounds_kernel<<<B_N, 256, 0, stream>>>(cc, segs, sege);

  dim3 grid(T_N / TILE_T, B_N, 1);
  seg_mean_kernel<<<grid, dim3(256, 1, 1), 0, stream>>>(H, segs, sege, out);
}